// DifferentiableCBFLayer_3805341024665
// MI455X (gfx1250) — compile-verified
//
#include <hip/hip_runtime.h>
#include <stdint.h>

// ---- problem constants (match reference) ----
#define NT      128      // threads per block (4 wave32)
#define MAXOBS  16
#define NAG     8
#define NITER   100
// D_OBS^2 = D_SAFE^2 = 0.25, D_MAX^2 = 100, DAMPING=2, STIFFNESS=1, W_SLACK=100, RHO=1

// ---- LDS slot layout: slot-major, thread-minor (conflict-free reads) ----
#define SLOT_U   0    // 2 slots
#define SLOT_V   2    // 1
#define SLOT_PO  3    // 32
#define SLOT_PA  35   // 16
#define SLOT_VA  51   // 16
#define SLOT_AA  67   // 8
#define SLOT_OA  75   // 16
#define NSLOT    91

// One async b32 transfer per lane: copies 4 bytes from the global address base + goff to the LDS address in lds_addr.
// Tracked on ASYNCcnt (CDNA5 async global->LDS path), no dest VGPRs held.
__device__ __forceinline__ void async_ld(uint32_t lds_addr, uint32_t goff, const float* base)
{
    asm volatile("global_load_async_to_lds_b32 %0, %1, %2"
                 :: "v"(lds_addr), "v"(goff), "s"(base)
                 : "memory");
}

template<int W>
__device__ __forceinline__ void stage(uint32_t sbase, int tid, int b, const float* g, int slot)
{
#pragma unroll
    for (int i = 0; i < W; ++i) {
        uint32_t la = sbase + (uint32_t)(((slot + i) * NT + tid) * 4);
        uint32_t go = (uint32_t)((b * W + i) * 4);
        async_ld(la, go, g);
    }
}

__global__ __launch_bounds__(NT)
void cbf_admm_kernel(const float* __restrict__ u_nom, const float* __restrict__ v_cur,
                     const float* __restrict__ p_obs, const float* __restrict__ p_ag,
                     const float* __restrict__ v_ag,  const float* __restrict__ a_act,
                     const float* __restrict__ o_act, float* __restrict__ out, int Btot)
{
    __shared__ float smem[NSLOT * NT];   // 46,592 B
    const int tid = threadIdx.x;
    const int b   = blockIdx.x * NT + tid;
    if (b >= Btot) return;

    // addrspace(3) offset of the LDS buffer (low 32 bits of flat LDS address)
    uint32_t sbase = (uint32_t)(uintptr_t)(&smem[0]);

    // ---- async stage: 91 in-flight async loads, transposed into LDS ----
    stage<2 >(sbase, tid, b, u_nom, SLOT_U);
    stage<1 >(sbase, tid, b, v_cur, SLOT_V);
    stage<32>(sbase, tid, b, p_obs, SLOT_PO);
    stage<16>(sbase, tid, b, p_ag,  SLOT_PA);
    stage<16>(sbase, tid, b, v_ag,  SLOT_VA);
    stage<8 >(sbase, tid, b, a_act, SLOT_AA);
    stage<16>(sbase, tid, b, o_act, SLOT_OA);
    asm volatile("s_wait_asynccnt 0x0" ::: "memory");
    // each wave stages and reads only its own lanes' region -> no barrier needed

#define SM(s) smem[(s) * NT + tid]

    const float u0 = SM(SLOT_U + 0);
    const float u1 = SM(SLOT_U + 1);
    const float vc = SM(SLOT_V);

    // ---- build constraint rows (loop-invariant) ----
    float c0[MAXOBS], c1[MAXOBS], bo[MAXOBS];          // obstacle rows, a2 = -1
#pragma unroll
    for (int o = 0; o < MAXOBS; ++o) {
        float lx  = SM(SLOT_PO + 2 * o);
        float ly  = SM(SLOT_PO + 2 * o + 1);
        float act = SM(SLOT_OA + o);
        c0[o] = 2.f * lx;
        c1[o] = 2.f * ly * vc;
        float h   = lx * lx + ly * ly - 0.25f;
        float hd  = -2.f * lx * vc;
        bo[o] = (2.f * vc * vc + 2.f * hd + h) * act;
    }

    float d0[NAG], d1[NAG], ba[NAG], bc[NAG];          // avoid rows; conn rows = -avoid
#pragma unroll
    for (int a = 0; a < NAG; ++a) {
        float lx  = SM(SLOT_PA + 2 * a);
        float ly  = SM(SLOT_PA + 2 * a + 1);
        float vjx = SM(SLOT_VA + 2 * a);
        float vjy = SM(SLOT_VA + 2 * a + 1);
        float act = SM(SLOT_AA + a);
        d0[a] = (2.f * lx) * act;
        d1[a] = (2.f * ly * vc - 2.f * ly * vjx + 2.f * lx * vjy) * act;
        float r2   = lx * lx + ly * ly;
        float h    = r2 - 0.25f;
        float hd   = -2.f * lx * vc + 2.f * (lx * vjx + ly * vjy);
        float hddc = 2.f * vc * vc - 4.f * vc * vjx + 2.f * (vjx * vjx + vjy * vjy);
        ba[a] = ( hddc + 2.f * hd + h) * act;
        bc[a] = (-hddc - 2.f * hd + (100.f - r2)) * act;
    }

    // ---- M = Q + A^T A (conn rows double avoid contributions; box rows constant) ----
    float M00 = 4.f, M01 = 0.f, M02 = 0.f, M11 = 4.f, M12 = 0.f;
    const float M22 = 217.f;   // 2*W_SLACK + 16 obstacle (-1)^2 + 1 box
#pragma unroll
    for (int o = 0; o < MAXOBS; ++o) {
        M00 += c0[o] * c0[o];  M01 += c0[o] * c1[o];  M02 -= c0[o];
        M11 += c1[o] * c1[o];  M12 -= c1[o];
    }
#pragma unroll
    for (int a = 0; a < NAG; ++a) {
        M00 += 2.f * d0[a] * d0[a];
        M01 += 2.f * d0[a] * d1[a];
        M11 += 2.f * d1[a] * d1[a];
    }
    // symmetric 3x3 inverse (cofactors)
    float C00 = M11 * M22 - M12 * M12;
    float C01 = M02 * M12 - M01 * M22;
    float C02 = M01 * M12 - M02 * M11;
    float rdet = 1.f / (M00 * C00 + M01 * C01 + M02 * C02);
    float i00 = C00 * rdet, i01 = C01 * rdet, i02 = C02 * rdet;
    float i11 = (M00 * M22 - M02 * M02) * rdet;
    float i12 = (M02 * M01 - M00 * M12) * rdet;
    float i22 = (M00 * M11 - M01 * M01) * rdet;

    // ---- ADMM state (rho=1): s = v + y ; z = min(s,b) ; y = max(s-b,0) ; t = min(s, 2b-s)
    float so[MAXOBS], sa[NAG], sc[NAG];
#pragma unroll
    for (int o = 0; o < MAXOBS; ++o) so[o] = fminf(0.f, bo[o]);
#pragma unroll
    for (int a = 0; a < NAG; ++a) { sa[a] = fminf(0.f, ba[a]); sc[a] = fminf(0.f, bc[a]); }
    float s32 = 0.f, s33 = 0.f, s34 = 0.f, s35 = 0.f, s36 = 0.f;  // box b=(1,1,1,1,0)>=0

    float ox0 = 0.f, ox1 = 0.f;
#pragma unroll 1
    for (int it = 0; it < NITER; ++it) {
        // rhs = -q + A^T t
        float r0 = 2.f * u0, r1 = 2.f * u1, r2 = 0.f;
#pragma unroll
        for (int o = 0; o < MAXOBS; ++o) {
            float t = fminf(so[o], 2.f * bo[o] - so[o]);
            r0 += c0[o] * t;  r1 += c1[o] * t;  r2 -= t;
        }
#pragma unroll
        for (int a = 0; a < NAG; ++a) {
            float t1 = fminf(sa[a], 2.f * ba[a] - sa[a]);
            float t2 = fminf(sc[a], 2.f * bc[a] - sc[a]);
            float td = t1 - t2;
            r0 += d0[a] * td;  r1 += d1[a] * td;
        }
        float t32 = fminf(s32, 2.f - s32);
        float t33 = fminf(s33, 2.f - s33);
        float t34 = fminf(s34, 2.f - s34);
        float t35 = fminf(s35, 2.f - s35);
        float t36 = fminf(s36, -s36);
        r0 += t33 - t32;  r1 += t35 - t34;  r2 -= t36;

        // x = Minv * rhs
        float x0 = i00 * r0 + i01 * r1 + i02 * r2;
        float x1 = i01 * r0 + i11 * r1 + i12 * r2;
        float x2 = i02 * r0 + i12 * r1 + i22 * r2;

        // v = A x ; s = v + max(s - b, 0)
#pragma unroll
        for (int o = 0; o < MAXOBS; ++o) {
            float v = c0[o] * x0 + c1[o] * x1 - x2;
            so[o] = v + fmaxf(so[o] - bo[o], 0.f);
        }
#pragma unroll
        for (int a = 0; a < NAG; ++a) {
            float vA = d0[a] * x0 + d1[a] * x1;
            sa[a] =  vA + fmaxf(sa[a] - ba[a], 0.f);
            sc[a] = -vA + fmaxf(sc[a] - bc[a], 0.f);
        }
        s32 = -x0 + fmaxf(s32 - 1.f, 0.f);
        s33 =  x0 + fmaxf(s33 - 1.f, 0.f);
        s34 = -x1 + fmaxf(s34 - 1.f, 0.f);
        s35 =  x1 + fmaxf(s35 - 1.f, 0.f);
        s36 = -x2 + fmaxf(s36, 0.f);

        ox0 = x0;  ox1 = x1;
    }

    // action_scale = (1,1) -> u_safe = x[:2]
    out[2 * b + 0] = ox0;
    out[2 * b + 1] = ox1;
#undef SM
}

extern "C" void kernel_launch(void* const* d_in, const int* in_sizes, int n_in,
                              void* d_out, int out_size, void* d_ws, size_t ws_size,
                              hipStream_t stream)
{
    const float* u_nom = (const float*)d_in[0];   // (B,2)
    const float* v_cur = (const float*)d_in[1];   // (B,1)
    const float* p_obs = (const float*)d_in[2];   // (B,16,2)
    const float* p_ag  = (const float*)d_in[3];   // (B,8,2)
    const float* v_ag  = (const float*)d_in[4];   // (B,8,2)
    const float* a_act = (const float*)d_in[5];   // (B,8)
    const float* o_act = (const float*)d_in[6];   // (B,16)
    float* out = (float*)d_out;

    const int B = in_sizes[0] / 2;
    dim3 grid((B + NT - 1) / NT), block(NT);
    cbf_admm_kernel<<<grid, block, 0, stream>>>(u_nom, v_cur, p_obs, p_ag, v_ag,
                                                a_act, o_act, out, B);
}